// WindowLayer_65833258713749
// MI455X (gfx1250) — compile-verified
//
#include <hip/hip_runtime.h>

typedef __attribute__((ext_vector_type(2))) float v2f;
typedef __attribute__((ext_vector_type(8))) float v8f;

#define BATCH   4096
#define INF     400
#define OUTF    400
#define NCOMP   10
#define NCHARS  80
#define SEQL    1024
#define CHUNK   256   // t-chunk held in LDS

// ---------------------------------------------------------------------------
// Kernel 1: o = exp(inputs @ W1^T + b1); split into a / bw / k; new_hidden out
// One thread per (row, component) pair: 4096*30 threads, 400-MAC dot each.
// ---------------------------------------------------------------------------
__global__ __launch_bounds__(256) void wl_mixture(
    const float* __restrict__ inp, const float* __restrict__ hidden,
    const float* __restrict__ W1, const float* __restrict__ b1,
    float* __restrict__ wsA, float* __restrict__ wsB, float* __restrict__ wsK,
    float* __restrict__ outH) {
  int idx = blockIdx.x * blockDim.x + threadIdx.x;
  if (idx >= BATCH * 3 * NCOMP) return;
  int row = idx / (3 * NCOMP);
  int c   = idx % (3 * NCOMP);
  const float* irow = inp + row * INF;
  const float* wrow = W1 + c * INF;
  float s = b1[c];
  for (int j = 0; j < INF; ++j) s = fmaf(irow[j], wrow[j], s);
  float o = __expf(s);
  if (c < NCOMP) {
    wsA[row * NCOMP + c] = o;
  } else if (c < 2 * NCOMP) {
    wsB[row * NCOMP + (c - NCOMP)] = o;
  } else {
    float nh = hidden[row * NCOMP + (c - 2 * NCOMP)] + o;
    wsK[row * NCOMP + (c - 2 * NCOMP)]  = nh;   // kappa used by phi
    outH[row * NCOMP + (c - 2 * NCOMP)] = nh;   // second tuple output
  }
}

// ---------------------------------------------------------------------------
// Kernel 2 (fused): per 16-row tile
//   phi chunk (LDS) -> window = phi @ seq (WMMA f32 16x16x4)
//                   -> out = window @ W2^T + b2 (WMMA f32 16x16x4)
// blockDim = 256 (8 waves of 32). grid = 4096/16 = 256 blocks.
// ---------------------------------------------------------------------------
__global__ __launch_bounds__(256) void wl_window(
    const float* __restrict__ seq, const float* __restrict__ W2,
    const float* __restrict__ b2,
    const float* __restrict__ wsA, const float* __restrict__ wsB,
    const float* __restrict__ wsK, float* __restrict__ out) {
  __shared__ float abk[3 * 16 * NCOMP];        // a | bw | k for 16 rows
  __shared__ float phi_lds[16 * CHUNK];        // 16 KB phi chunk
  __shared__ float win_lds[16 * NCHARS];       // 16x80 window tile

  const int tid  = threadIdx.x;
  const int r0   = blockIdx.x * 16;
  const int wave = tid >> 5;
  const int lane = tid & 31;
  const int mn    = lane & 15;                 // M for A-frag, N for B/C-frag
  const int basek = (lane >> 4) << 1;          // K pairing: 0 (lanes 0-15) / 2 (16-31)
  const int rowhi = (lane >> 4) << 3;          // C/D: +8 rows for upper half-wave

  for (int i = tid; i < 16 * NCOMP; i += 256) {
    abk[i]                  = wsA[r0 * NCOMP + i];
    abk[16 * NCOMP + i]     = wsB[r0 * NCOMP + i];
    abk[32 * NCOMP + i]     = wsK[r0 * NCOMP + i];
  }
  __syncthreads();

  v8f accw = {};  // window accumulator (waves 0..4, one 16-col N-tile each)

  for (int t0 = 0; t0 < SEQL; t0 += CHUNK) {
    // ---- phase A: thread `tid` computes phi[:, t0+tid] for all 16 rows ----
    {
      const float tt = (float)(t0 + tid);
      for (int m = 0; m < 16; ++m) {
        float s = 0.f;
        #pragma unroll
        for (int n = 0; n < NCOMP; ++n) {
          float a  = abk[m * NCOMP + n];
          float bw = abk[16 * NCOMP + m * NCOMP + n];
          float d  = abk[32 * NCOMP + m * NCOMP + n] - tt;
          s = fmaf(a, __expf(-bw * d * d), s);
        }
        phi_lds[m * CHUNK + tid] = s;
      }
    }
    __syncthreads();

    // ---- phase B: window += phi_chunk @ seq_chunk (waves 0..4) ----
    if (wave < 5) {
      const int c0 = wave * 16;
      for (int kk = 0; kk < CHUNK; kk += 4) {
        v2f af, bf;
        af.x = phi_lds[mn * CHUNK + kk + basek];
        af.y = phi_lds[mn * CHUNK + kk + basek + 1];
        const int t = t0 + kk + basek;
        bf.x = seq[t * NCHARS + c0 + mn];
        bf.y = seq[(t + 1) * NCHARS + c0 + mn];
        accw = __builtin_amdgcn_wmma_f32_16x16x4_f32(
            false, af, false, bf, (short)0, accw, false, false);
      }
    }
    __syncthreads();
  }

  if (wave < 5) {
    const int c0 = wave * 16;
    #pragma unroll
    for (int r = 0; r < 8; ++r)
      win_lds[(r + rowhi) * NCHARS + c0 + mn] = accw[r];
  }
  __syncthreads();

  // ---- phase C: out = window @ W2^T + b2; 25 N-tiles over 8 waves ----
  for (int tix = wave; tix < OUTF / 16; tix += 8) {
    const int n0 = tix * 16;
    v8f acco = {};
    for (int k0 = 0; k0 < NCHARS; k0 += 4) {
      v2f af, bf;
      af.x = win_lds[mn * NCHARS + k0 + basek];
      af.y = win_lds[mn * NCHARS + k0 + basek + 1];
      const float* w2r = W2 + (n0 + mn) * NCHARS + k0 + basek; // B[k][n] = W2[n][k]
      bf.x = w2r[0];
      bf.y = w2r[1];
      acco = __builtin_amdgcn_wmma_f32_16x16x4_f32(
          false, af, false, bf, (short)0, acco, false, false);
    }
    const float bias = b2[n0 + mn];
    #pragma unroll
    for (int r = 0; r < 8; ++r)
      out[(r0 + r + rowhi) * OUTF + n0 + mn] = acco[r] + bias;
  }
}

extern "C" void kernel_launch(void* const* d_in, const int* in_sizes, int n_in,
                              void* d_out, int out_size, void* d_ws, size_t ws_size,
                              hipStream_t stream) {
  const float* inp    = (const float*)d_in[0];
  const float* hidden = (const float*)d_in[1];
  const float* seq    = (const float*)d_in[2];
  const float* W1     = (const float*)d_in[3];
  const float* b1     = (const float*)d_in[4];
  const float* W2     = (const float*)d_in[5];
  const float* b2     = (const float*)d_in[6];

  float* out  = (float*)d_out;            // (4096, 400) flat
  float* outH = out + BATCH * OUTF;       // (4096, 10) flat, second tuple member

  float* wsA = (float*)d_ws;              // a  (4096x10)
  float* wsB = wsA + BATCH * NCOMP;       // bw (4096x10)
  float* wsK = wsB + BATCH * NCOMP;       // k  (4096x10)

  const int tot = BATCH * 3 * NCOMP;
  wl_mixture<<<(tot + 255) / 256, 256, 0, stream>>>(inp, hidden, W1, b1,
                                                    wsA, wsB, wsK, outH);
  wl_window<<<BATCH / 16, 256, 0, stream>>>(seq, W2, b2, wsA, wsB, wsK, out);
}